// RelaxBoundaryLoss_7017976561935
// MI455X (gfx1250) — compile-verified
//
#include <hip/hip_runtime.h>
#include <hip/hip_bf16.h>
#include <cstdint>

// Problem constants (reference: N,C,H,W = 4,19,768,768)
#define IMG_W 768
#define IMG_H 768
#define NCLS  19
#define NIMG  4
#define TX    32
#define TY    8
#define TILE  (TX * TY)                 // 256 threads / block (8 wave32)
#define BX    (IMG_W / TX)              // 24
#define BY    (IMG_H / TY)              // 96
#define BLKS_PER_IMG (BX * BY)          // 2304
#define CHUNKS (NCLS * TILE / 4)        // 1216 x 16B chunks = full logit tile

__device__ __forceinline__ float wave_reduce_add(float v) {
    // wave32 tree reduction; lane 0 ends up with the sum
    v += __shfl_down(v, 16, 32);
    v += __shfl_down(v, 8, 32);
    v += __shfl_down(v, 4, 32);
    v += __shfl_down(v, 2, 32);
    v += __shfl_down(v, 1, 32);
    return v;
}

// OR of class bits over the 3 vertical neighbors of (xx, y); out-of-image or
// IGNORE(255)/pad contribute nothing (== class 19 column that reference drops).
__device__ __forceinline__ uint32_t col_mask(const int* __restrict__ lb, int xx, int y) {
    uint32_t m = 0u;
    if (xx >= 0 && xx < IMG_W) {
#pragma unroll
        for (int dy = -1; dy <= 1; ++dy) {
            const int yy = y + dy;
            if (yy >= 0 && yy < IMG_H) {
                const uint32_t c = (uint32_t)lb[(size_t)yy * IMG_W + xx];
                if (c < NCLS) m |= (1u << c);
            }
        }
    }
    return m;
}

__global__ __launch_bounds__(TILE) void relax_loss_main(
    const float* __restrict__ logit,   // [N, C, H, W]
    const int*   __restrict__ label,   // [N, H, W]
    float* __restrict__ part_loss,     // [N * BLKS_PER_IMG]
    float* __restrict__ part_ign)      // [N * BLKS_PER_IMG]
{
    __shared__ float    lds_logit[NCLS * TILE];   // 19 KB: [class][pixel]
    __shared__ uint32_t cm[TY][TX + 2];           // column masks + x halo
    __shared__ float    wsum[TILE / 32];
    __shared__ float    wign[TILE / 32];

    const int tid = threadIdx.x;
    const int lx  = tid & (TX - 1);
    const int ly  = tid >> 5;
    const int x0  = blockIdx.x * TX;
    const int y0  = blockIdx.y * TY;
    const int x   = x0 + lx;
    const int y   = y0 + ly;
    const int n   = blockIdx.z;

    const size_t plane = (size_t)IMG_H * IMG_W;

    // ---- Cooperatively stage the whole 19-class logit tile into LDS with
    //      gfx1250 async global->LDS b128 chunks (ASYNCcnt): 1216 chunks of
    //      16B per block, ~4.75 VMEM ops/thread for the entire hot traffic.
    {
        const float*   gbase    = logit + (size_t)n * NCLS * plane
                                        + (size_t)y0 * IMG_W + x0;
        const uint32_t lds_base = (uint32_t)(uintptr_t)lds_logit;
        for (int i = tid; i < CHUNKS; i += TILE) {
            const int c = i >> 6;          // class (64 chunks per class)
            const int r = i & 63;          // 4-pixel group within tile
            const float* gp = gbase + (size_t)c * plane
                                    + (size_t)(r >> 3) * IMG_W + ((r & 7) << 2);
            const uint32_t loff = lds_base + (uint32_t)i * 16u;
            asm volatile("global_load_async_to_lds_b128 %0, %1, off"
                         :: "v"(loff), "v"(gp)
                         : "memory");
        }
    }

    // ---- Labels while the DMA flies: per-thread column mask (3 loads),
    //      edge lanes add the two x-halo columns; 3x3 OR becomes 3 LDS ORs.
    {
        const int* lb = label + (size_t)n * plane;
        cm[ly][lx + 1] = col_mask(lb, x, y);
        if (lx == 0)      cm[ly][0]      = col_mask(lb, x - 1, y);
        if (lx == TX - 1) cm[ly][TX + 1] = col_mask(lb, x + 1, y);
    }

    // Each wave drains its own async LDS writes, then the block barrier makes
    // every wave's staged chunks (and column masks) visible to all.
    asm volatile("s_wait_asynccnt 0x0" ::: "memory");
    __syncthreads();

    const uint32_t mask = cm[ly][lx] | cm[ly][lx + 1] | cm[ly][lx + 2];

    // ---- Per-pixel stable softmax-set log-prob, values held in registers.
    float v[NCLS];
#pragma unroll
    for (int c = 0; c < NCLS; ++c)
        v[c] = lds_logit[c * TILE + tid];

    float m = v[0];
#pragma unroll
    for (int c = 1; c < NCLS; ++c)
        m = fmaxf(m, v[c]);

    float se = 0.0f, ss = 0.0f;
#pragma unroll
    for (int c = 0; c < NCLS; ++c) {
        const float e = __builtin_amdgcn_exp2f((v[c] - m) * 1.4426950408889634f);
        se += e;
        ss += ((mask >> c) & 1u) ? e : 0.0f;
    }

    // per_px = -log(P_S) = (log2(se) - log2(ss)) * ln2 ; ignored pixels -> 0
    float per = 0.0f, ign = 0.0f;
    if (mask != 0u) {
        per = (__builtin_amdgcn_logf(se) - __builtin_amdgcn_logf(ss))
              * 0.69314718055994531f;
    } else {
        ign = 1.0f;
    }

    // ---- Deterministic block reduction (wave32 shuffles, fixed order).
    float s  = wave_reduce_add(per);
    float gi = wave_reduce_add(ign);
    const int lane = tid & 31;
    const int wv   = tid >> 5;
    if (lane == 0) { wsum[wv] = s; wign[wv] = gi; }
    __syncthreads();
    if (wv == 0) {
        float a = (lane < TILE / 32) ? wsum[lane] : 0.0f;
        float b = (lane < TILE / 32) ? wign[lane] : 0.0f;
        a = wave_reduce_add(a);
        b = wave_reduce_add(b);
        if (lane == 0) {
            const int bid = blockIdx.y * BX + blockIdx.x;
            part_loss[(size_t)n * BLKS_PER_IMG + bid] = a;
            part_ign [(size_t)n * BLKS_PER_IMG + bid] = b;
        }
    }
}

__global__ __launch_bounds__(32) void relax_loss_final(
    const float* __restrict__ part_loss,
    const float* __restrict__ part_ign,
    float* __restrict__ out)
{
    const int lane = threadIdx.x;   // single wave32, fixed-order reduction
    float loss = 0.0f;
    for (int n = 0; n < NIMG; ++n) {
        float s = 0.0f, gi = 0.0f;
        for (int i = lane; i < BLKS_PER_IMG; i += 32) {
            s  += part_loss[n * BLKS_PER_IMG + i];
            gi += part_ign [n * BLKS_PER_IMG + i];
        }
        s  = wave_reduce_add(s);
        gi = wave_reduce_add(gi);
        if (lane == 0) {
            const float denom = (float)(IMG_H * IMG_W) - gi + 1.0f;
            loss += s / denom;
        }
    }
    if (lane == 0) out[0] = loss;
}

extern "C" void kernel_launch(void* const* d_in, const int* in_sizes, int n_in,
                              void* d_out, int out_size, void* d_ws, size_t ws_size,
                              hipStream_t stream) {
    const float* logit = (const float*)d_in[0];   // [4,19,768,768] f32
    const int*   label = (const int*)d_in[1];     // [4,768,768] int

    float* part_loss = (float*)d_ws;                        // 9216 floats
    float* part_ign  = part_loss + NIMG * BLKS_PER_IMG;     // 9216 floats (72 KB total)

    dim3 grid(BX, BY, NIMG);
    relax_loss_main<<<grid, TILE, 0, stream>>>(logit, label, part_loss, part_ign);
    relax_loss_final<<<1, 32, 0, stream>>>(part_loss, part_ign, (float*)d_out);
}